// PoseAttRouter_90022514524280
// MI455X (gfx1250) — compile-verified
//
#include <hip/hip_runtime.h>
#include <hip/hip_bf16.h>
#include <math.h>

typedef __attribute__((ext_vector_type(2))) float v2f;
typedef __attribute__((ext_vector_type(8))) float v8f;

#define B_DIM 64
#define N_DIM 1152
#define C_DIM 32
#define D_DIM 16

#if __has_builtin(__builtin_amdgcn_global_load_async_to_lds_b128) && \
    __has_builtin(__builtin_amdgcn_s_wait_asynccnt)
#define USE_ASYNC_LDS 1
#else
#define USE_ASYNC_LDS 0
#endif

#if USE_ASYNC_LDS
// Exact pointee type clang expects for the b128 async builtin:
// "__attribute__((__vector_size__(4 * sizeof(int)))) int" in AS(1)/AS(3).
typedef int a_v4i __attribute__((vector_size(16)));
typedef __attribute__((address_space(1))) a_v4i* gptr_v4i;
typedef __attribute__((address_space(3))) a_v4i* lptr_v4i;
#endif

// ---------------------------------------------------------------------------
// Kernel 1: M[b,c,p,q] = sum_n A[b,n,c] * Q[b,n,c,p] * Q[b,n,c,q]
// One workgroup per b (256 thr = 8 waves). Wave w handles channel group w
// (channels 4w..4w+3) via V_WMMA_F32_16X16X4_F32. Double-buffered LDS tiles
// staged with GLOBAL_LOAD_ASYNC_TO_LDS_B128 (ASYNCcnt) so the next 20 KB
// chunk's HBM latency overlaps the WMMA chain on the current chunk.
// ---------------------------------------------------------------------------
__global__ void __launch_bounds__(256)
k1_moments(const float* __restrict__ A, const float* __restrict__ Q,
           float* __restrict__ wsM) {
    constexpr int NT  = 32;                      // n-chunk per LDS stage
    constexpr int NCH = N_DIM / NT;              // 36 chunks
    const int b    = blockIdx.x;
    const int tid  = threadIdx.x;
    const int wv   = tid >> 5;                   // wave id == channel group
    const int lane = tid & 31;

    __shared__ float sA[2][NT * C_DIM];          // 2 x 4 KB
    __shared__ float sQ[2][NT * C_DIM * 4];      // 2 x 16 KB

    const int half = lane >> 4;                  // 0: K={0,1}, 1: K={2,3}
    const int idx  = lane & 15;                  // A-row / B-col index
    const int c_l  = idx >> 2;
    const int pq   = idx & 3;
    const int cch  = (wv << 2) + c_l;            // channel 0..31

    v8f acc = {0.f, 0.f, 0.f, 0.f, 0.f, 0.f, 0.f, 0.f};

    const float4* A4 = reinterpret_cast<const float4*>(A);
    const float4* Q4 = reinterpret_cast<const float4*>(Q);

    // Issue one chunk's staging loads: per thread 1 A-float4 + 4 Q-float4.
    auto issue = [&](int buf, int n0) {
        const size_t abase = (size_t)(b * N_DIM + n0) * (C_DIM / 4);
        const size_t qbase = (size_t)(b * N_DIM + n0) * C_DIM;   // float4 units
#if USE_ASYNC_LDS
        __builtin_amdgcn_global_load_async_to_lds_b128(
            (gptr_v4i)(A4 + abase + tid),
            (lptr_v4i)(&sA[buf][4 * tid]), 0, 0);
        #pragma unroll
        for (int j = 0; j < 4; ++j)
            __builtin_amdgcn_global_load_async_to_lds_b128(
                (gptr_v4i)(Q4 + qbase + tid + 256 * j),
                (lptr_v4i)(&sQ[buf][4 * (tid + 256 * j)]), 0, 0);
#else
        reinterpret_cast<float4*>(sA[buf])[tid] = A4[abase + tid];
        #pragma unroll
        for (int j = 0; j < 4; ++j)
            reinterpret_cast<float4*>(sQ[buf])[tid + 256 * j] =
                Q4[qbase + tid + 256 * j];
#endif
    };

    issue(0, 0);                                 // prologue: fill buffer 0

    for (int ci = 0; ci < NCH; ++ci) {
        const int buf = ci & 1;
        if (ci + 1 < NCH) {
            issue(buf ^ 1, (ci + 1) * NT);       // prefetch next chunk
#if USE_ASYNC_LDS
            __builtin_amdgcn_s_wait_asynccnt(5); // only next chunk's 5 in flight
#endif
        } else {
#if USE_ASYNC_LDS
            __builtin_amdgcn_s_wait_asynccnt(0); // drain: current chunk ready
#endif
        }
        __syncthreads();                         // all waves see current buffer

        const float* cA = sA[buf];
        const float* cQ = sQ[buf];
        #pragma unroll
        for (int kk = 0; kk < NT / 4; ++kk) {
            const int nk = kk * 4 + 2 * half;    // this lane's first K slice
            const float q0 = cQ[(nk    ) * 128 + cch * 4 + pq];
            const float q1 = cQ[(nk + 1) * 128 + cch * 4 + pq];
            const float w0 = cA[(nk    ) * 32  + cch];
            const float w1 = cA[(nk + 1) * 32  + cch];
            v2f av = {w0 * q0, w1 * q1};         // A fragment (weighted Q)
            v2f bv = {q0, q1};                   // B fragment (plain Q)
            acc = __builtin_amdgcn_wmma_f32_16x16x4_f32(
                false, av, false, bv, (short)0, acc, false, false);
        }
        __syncthreads();                         // compute done before overwrite
    }

    // Store full 16x16 D; kernel 2 picks diagonal 4x4 blocks.
    float* dst = wsM + (size_t)(b * 8 + wv) * 256;
    #pragma unroll
    for (int r = 0; r < 8; ++r)
        dst[(r + 8 * half) * 16 + idx] = acc[r];
}

// ---------------------------------------------------------------------------
// Kernel 2: 4x4 symmetric eigendecomposition (cyclic Jacobi, unrolled),
// one thread per (b,c). Produces lam_max/sum(lam) and K = last row of the
// eigenvector matrix with eigenvalues sorted ascending (matches reference).
// ---------------------------------------------------------------------------
template <int P, int Qq>
__device__ inline void jrot(float (&a)[4][4], float (&v)[4][4]) {
    const float apq = a[P][Qq];
    if (fabsf(apq) < 1e-18f) return;
    const float app = a[P][P], aqq = a[Qq][Qq];
    const float tau = (aqq - app) / (2.0f * apq);
    const float t   = (tau >= 0.0f ? 1.0f : -1.0f) /
                      (fabsf(tau) + sqrtf(1.0f + tau * tau));
    const float cth = 1.0f / sqrtf(1.0f + t * t);
    const float sth = t * cth;
    #pragma unroll
    for (int k = 0; k < 4; ++k) {
        const float akp = a[k][P], akq = a[k][Qq];
        a[k][P]  = cth * akp - sth * akq;
        a[k][Qq] = sth * akp + cth * akq;
    }
    #pragma unroll
    for (int k = 0; k < 4; ++k) {
        const float apk = a[P][k], aqk = a[Qq][k];
        a[P][k]  = cth * apk - sth * aqk;
        a[Qq][k] = sth * apk + cth * aqk;
    }
    #pragma unroll
    for (int k = 0; k < 4; ++k) {
        const float vkp = v[k][P], vkq = v[k][Qq];
        v[k][P]  = cth * vkp - sth * vkq;
        v[k][Qq] = sth * vkp + cth * vkq;
    }
}

__global__ void __launch_bounds__(256)
k2_eig(const float* __restrict__ wsM, float* __restrict__ lamfrac,
       float* __restrict__ out_q) {
    const int g = blockIdx.x * blockDim.x + threadIdx.x;   // 0..2047
    if (g >= B_DIM * C_DIM) return;
    const int b = g >> 5, c = g & 31;

    const float* Mp = wsM + (size_t)(b * 8 + (c >> 2)) * 256;
    const int off = (c & 3) * 4;

    float a[4][4], v[4][4];
    #pragma unroll
    for (int p = 0; p < 4; ++p)
        #pragma unroll
        for (int q = 0; q < 4; ++q) {
            a[p][q] = Mp[(off + p) * 16 + (off + q)];
            v[p][q] = (p == q) ? 1.0f : 0.0f;
        }

    #pragma unroll
    for (int sweep = 0; sweep < 8; ++sweep) {
        jrot<0, 1>(a, v); jrot<0, 2>(a, v); jrot<0, 3>(a, v);
        jrot<1, 2>(a, v); jrot<1, 3>(a, v); jrot<2, 3>(a, v);
    }

    float lam[4] = {a[0][0], a[1][1], a[2][2], a[3][3]};
    float kr[4]  = {v[3][0], v[3][1], v[3][2], v[3][3]};  // last row of eigvects

    #define CSWAP(i, j)                                        \
        if (lam[i] > lam[j]) {                                 \
            float tl = lam[i]; lam[i] = lam[j]; lam[j] = tl;   \
            float tk = kr[i];  kr[i]  = kr[j];  kr[j]  = tk;   \
        }
    CSWAP(0, 1) CSWAP(2, 3) CSWAP(0, 2) CSWAP(1, 3) CSWAP(1, 2)
    #undef CSWAP

    const float s = lam[0] + lam[1] + lam[2] + lam[3];
    lamfrac[g] = lam[3] / s;
    out_q[g * 4 + 0] = kr[0];
    out_q[g * 4 + 1] = kr[1];
    out_q[g * 4 + 2] = kr[2];
    out_q[g * 4 + 3] = kr[3];
}

// ---------------------------------------------------------------------------
// Kernel 3: out_a = softmax over C of lam_max/sum(lam). One wave per b.
// ---------------------------------------------------------------------------
__global__ void __launch_bounds__(32)
k3_outa(const float* __restrict__ lamfrac, float* __restrict__ out_a) {
    __shared__ float s[C_DIM];
    const int b = blockIdx.x, c = threadIdx.x;
    const float vv = lamfrac[b * C_DIM + c];
    s[c] = vv;
    __syncthreads();
    float m = -1e30f;
    for (int i = 0; i < C_DIM; ++i) m = fmaxf(m, s[i]);
    __syncthreads();
    const float e = __expf(vv - m);
    s[c] = e;
    __syncthreads();
    float den = 0.f;
    for (int i = 0; i < C_DIM; ++i) den += s[i];
    out_a[b * C_DIM + c] = e / den;
}

// ---------------------------------------------------------------------------
// Kernel 4: attention pass. One block per b; all 1152x32 logits live in
// CDNA5's big LDS (148 KB < 320 KB/WGP), so Q is read exactly once here and
// V exactly once, fully coalesced. Wave32 shuffle reductions per channel.
// Speculative prefetch runs 16 rows ahead of the dominant V stream.
// ---------------------------------------------------------------------------
__global__ void __launch_bounds__(512)
k4_outf(const float* __restrict__ Q, const float* __restrict__ V,
        const float* __restrict__ Kq, float* __restrict__ out_f) {
    __shared__ float sE[N_DIM * C_DIM];   // 144 KB: logits -> exp weights
    __shared__ float sK[C_DIM * 4];
    __shared__ float sMax[C_DIM];
    __shared__ float sDen[C_DIM];

    const int b = blockIdx.x, tid = threadIdx.x;
    if (tid < C_DIM * 4) sK[tid] = Kq[b * C_DIM * 4 + tid];
    __syncthreads();

    // Phase 1: logits s = 1/dist = (pi/2) / acos(min(|Q.K|, 0.9999))
    const float4* Q4 = reinterpret_cast<const float4*>(Q) + (size_t)b * N_DIM * C_DIM;
    for (int i = tid; i < N_DIM * C_DIM; i += 512) {
        const int c = i & 31;
        const float4 qv = Q4[i];
        const float dot = qv.x * sK[c * 4 + 0] + qv.y * sK[c * 4 + 1] +
                          qv.z * sK[c * 4 + 2] + qv.w * sK[c * 4 + 3];
        const float ad = fminf(fabsf(dot), 0.9999f);
        sE[i] = 1.5707963267948966f / acosf(ad);
    }
    __syncthreads();

    // Phase 2: per-channel max & sum(exp) over N, one wave per channel pair.
    const int wv = tid >> 5, lane = tid & 31;
    for (int c = wv; c < C_DIM; c += 16) {
        float m = -1e30f;
        for (int n = lane; n < N_DIM; n += 32) m = fmaxf(m, sE[n * C_DIM + c]);
        #pragma unroll
        for (int o = 16; o > 0; o >>= 1) m = fmaxf(m, __shfl_xor(m, o, 32));
        float den = 0.f;
        for (int n = lane; n < N_DIM; n += 32) den += __expf(sE[n * C_DIM + c] - m);
        #pragma unroll
        for (int o = 16; o > 0; o >>= 1) den += __shfl_xor(den, o, 32);
        if (lane == 0) { sMax[c] = m; sDen[c] = den; }
    }
    __syncthreads();

    // Phase 2.5: logits -> unnormalized weights in LDS
    for (int i = tid; i < N_DIM * C_DIM; i += 512) {
        const int c = i & 31;
        sE[i] = __expf(sE[i] - sMax[c]);
    }
    __syncthreads();

    // Phase 3: out_f[c,d] = (1/N) * sum_n w[n,c]*V[n,c,d] / den[c]
    // tid spans (c,d): per-n reads of V are 512 contiguous floats.
    const int c = tid >> 4;
    const float* Vb = V + (size_t)b * N_DIM * C_DIM * D_DIM;
    float acc = 0.f;
    for (int n = 0; n < N_DIM; ++n) {
        __builtin_prefetch(&Vb[(size_t)(n + 16) * 512 + tid], 0, 0);
        acc = fmaf(sE[n * C_DIM + c], Vb[(size_t)n * 512 + tid], acc);
    }
    out_f[b * 512 + tid] = acc / (sDen[c] * (float)N_DIM);
}

// ---------------------------------------------------------------------------
extern "C" void kernel_launch(void* const* d_in, const int* in_sizes, int n_in,
                              void* d_out, int out_size, void* d_ws, size_t ws_size,
                              hipStream_t stream) {
    const float* A = (const float*)d_in[0];   // (64,1152,32)
    const float* Q = (const float*)d_in[1];   // (64,1152,32,4)
    const float* V = (const float*)d_in[2];   // (64,1152,32,16)

    float* out   = (float*)d_out;
    float* out_a = out;                       // (64,32,1)  = 2048
    float* out_q = out + 2048;                // (64,32,4)  = 8192
    float* out_f = out + 2048 + 8192;         // (64,32,16) = 32768

    float* wsM     = (float*)d_ws;            // 64*8*256 floats = 512 KB
    float* lamfrac = wsM + 64 * 8 * 256;      // 2048 floats

    k1_moments<<<B_DIM, 256, 0, stream>>>(A, Q, wsM);
    k2_eig<<<(B_DIM * C_DIM) / 256, 256, 0, stream>>>(wsM, lamfrac, out_q);
    k3_outa<<<B_DIM, C_DIM, 0, stream>>>(lamfrac, out_a);
    k4_outf<<<B_DIM, 512, 0, stream>>>(Q, V, out_q, out_f);
}